// Mooncake_16080357556690
// MI455X (gfx1250) — compile-verified
//
#include <hip/hip_runtime.h>
#include <math.h>

typedef __attribute__((ext_vector_type(2))) float v2f;
typedef __attribute__((ext_vector_type(8))) float v8f;

#define C_      64
#define TWO_C   128
#define T_FULL  4096
#define BATCH   64
#define TV      256          // valid outputs per tile
#define HALO    256          // receptive field
#define P_      512          // positions held in h buffer
#define HSTR    528          // h row stride in floats (528 % 64 == 16 -> bank-friendly)
#define ASTR    264          // acc / z row stride   (264 % 64 == 8)
#define NTILES  (T_FULL / TV)  // 16
#define NBLK    8
#define NW      8            // waves per workgroup (wave32)
#define NTHREADS 256

// Misc (small constants) offsets inside LDS
#define MB1   0      // blk_b1 (128)
#define MB2   128    // blk_b2 (64)
#define MWIN  192    // w_in (64*2*2)
#define MBIN  448    // b_in (64)
#define MBMID 512    // b_mid (128)
#define MWPI  640    // w_pi (4*128)
#define MWMU  1152   // w_mu (3*128)
#define MWSG  1536   // w_sigma (3*128)
#define MBPI  1920
#define MBMU  1924
#define MBSG  1928
#define MISC_SZ 2048

#define SMEM_FLOATS (64*HSTR + 64*ASTR + 16384 + 4096 + MISC_SZ)  // 73216 -> 292864 B

#define OFF_MU (BATCH*4*T_FULL)                 // 1048576
#define OFF_SG (OFF_MU + BATCH*3*T_FULL)        // 1835008

static __device__ __forceinline__ v8f wmma4(v2f a, v2f b, v8f c) {
  // D = A(16x4 f32) * B(4x16 f32) + C(16x16 f32)
  return __builtin_amdgcn_wmma_f32_16x16x4_f32(false, a, false, b, (short)0, c,
                                               false, false);
}

static __device__ __forceinline__ float sx16(float v) {
  return __shfl_xor(v, 16, 32);   // swap half-waves (wave32)
}

// hardware tanh (CDNA5 V_TANH_F32) when the builtin exists
#if __has_builtin(__builtin_amdgcn_tanhf)
#define HTANH(x) __builtin_amdgcn_tanhf((x))
#elif __has_builtin(__builtin_amdgcn_tanh_f32)
#define HTANH(x) __builtin_amdgcn_tanh_f32((x))
#else
#define HTANH(x) tanhf((x))
#endif

static __device__ __forceinline__ float fsigmoid(float x) {
  // v_exp + v_rcp instead of full-precision division
  return __builtin_amdgcn_rcpf(1.0f + __expf(-x));
}

// CDNA5 async global -> LDS copy (tracked by ASYNCcnt, bypasses VGPRs).
// Generic-pointer low 32 bits are the LDS byte address (ISA 10.2).
static __device__ __forceinline__ void async_g2l_b32(const float* lds_dst,
                                                     unsigned goff_bytes,
                                                     const float* gbase) {
  unsigned lds_addr = (unsigned)(unsigned long long)(const void*)lds_dst;
  asm volatile("global_load_async_to_lds_b32 %0, %1, %2"
               :: "v"(lds_addr), "v"(goff_bytes),
                  "s"((unsigned long long)(const void*)gbase)
               : "memory");
}
static __device__ __forceinline__ void async_wait0() {
#if __has_builtin(__builtin_amdgcn_s_wait_asynccnt)
  __builtin_amdgcn_s_wait_asynccnt(0);
#else
  asm volatile("s_wait_asynccnt 0" ::: "memory");
#endif
}

// write a staged z tile back to Hbuf and accumulate into Acc
#define FLUSH(ST, CK)                                                   \
  do {                                                                  \
    if ((CK) >= 0) {                                                    \
      int wcol = (CK) * 16 + ln;                                        \
      _Pragma("unroll")                                                 \
      for (int zr = 0; zr < 4; ++zr) {                                  \
        _Pragma("unroll")                                               \
        for (int rr = 0; rr < 8; ++rr) {                                \
          int ch = zr * 16 + rr + 8 * hf;                               \
          float vv = (ST)[zr][rr];                                      \
          Hbuf[ch * HSTR + wcol] = vv;                                  \
          if (wcol >= HALO) Acc[ch * ASTR + (wcol - HALO)] += vv;       \
        }                                                               \
      }                                                                 \
      (CK) = -1;                                                        \
    }                                                                   \
  } while (0)

__global__ __launch_bounds__(NTHREADS, 1)
void Mooncake_wavenet_kernel(const float* __restrict__ x,
                             const float* __restrict__ w_in,  const float* __restrict__ b_in,
                             const float* __restrict__ blk_w1,const float* __restrict__ blk_b1,
                             const float* __restrict__ blk_w2,const float* __restrict__ blk_b2,
                             const float* __restrict__ w_mid, const float* __restrict__ b_mid,
                             const float* __restrict__ w_pi,  const float* __restrict__ b_pi,
                             const float* __restrict__ w_mu,  const float* __restrict__ b_mu,
                             const float* __restrict__ w_sg,  const float* __restrict__ b_sg,
                             float* __restrict__ out)
{
  extern __shared__ float smem[];
  float* Hbuf = smem;                  // [64][HSTR]  (reused as Z: [128][ASTR])
  float* Acc  = Hbuf + 64 * HSTR;      // [64][ASTR]
  float* W1s  = Acc  + 64 * ASTR;      // [tap][o=128][c=64]  (also w_mid in epilogue)
  float* W2s  = W1s + 16384;           // [o=64][c=64]
  float* Misc = W2s + 4096;

  const int tid  = threadIdx.x;
  const int wv   = tid >> 5;
  const int lane = tid & 31;
  const int hf   = lane >> 4;          // which half-wave (K/row group)
  const int ln   = lane & 15;          // row (A) / column (B,C,D) index

  const int b    = blockIdx.x / NTILES;
  const int tile = blockIdx.x % NTILES;
  const int t0   = tile * TV;

  const float* xg = x + b * T_FULL;

  // ---------------- Phase 0: one-time small constants -> LDS ----------------
  for (int i = tid; i < 128; i += NTHREADS) Misc[MBMID + i] = b_mid[i];
  for (int i = tid; i < 256; i += NTHREADS) Misc[MWIN  + i] = w_in[i];
  for (int i = tid; i < 64;  i += NTHREADS) Misc[MBIN  + i] = b_in[i];
  for (int i = tid; i < 512; i += NTHREADS) Misc[MWPI  + i] = w_pi[i];
  for (int i = tid; i < 384; i += NTHREADS) Misc[MWMU  + i] = w_mu[i];
  for (int i = tid; i < 384; i += NTHREADS) Misc[MWSG  + i] = w_sg[i];
  if (tid < 4) Misc[MBPI + tid] = b_pi[tid];
  if (tid < 3) Misc[MBMU + tid] = b_mu[tid];
  if (tid < 3) Misc[MBSG + tid] = b_sg[tid];
  __syncthreads();

  // ---------------- Phase 1: input conv (2 -> 64 ch) into Hbuf, init Acc ----
  const int pstart = (tile == 0) ? HALO : 0;
  if (tile == 0) {  // global t < 0 must be exactly zero (zero-padding semantics)
    for (int i = tid; i < 64 * HALO; i += NTHREADS)
      Hbuf[(i & 63) * HSTR + (i >> 6)] = 0.0f;
  }
  {
    const float inv_lin = 1.0f / 4095.0f;  // linspace(0,1,4096) step
    const int npos = P_ - pstart;
    for (int i = tid; i < 64 * npos; i += NTHREADS) {
      int ch = i & 63;
      int pp = pstart + (i >> 6);
      int t  = t0 - HALO + pp;             // >= 0 by construction
      float x0  = xg[t];
      float xm1 = (t >= 1) ? xg[t - 1] : 0.0f;
      float l0  = (float)t * inv_lin;
      float lm1 = (t >= 1) ? (float)(t - 1) * inv_lin : 0.0f;
      const float* wc = &Misc[MWIN + ch * 4];  // [c][in=2][k=2]
      float h = wc[0] * xm1 + wc[1] * x0 + wc[2] * lm1 + wc[3] * l0 + Misc[MBIN + ch];
      Hbuf[ch * HSTR + pp] = h;
      if (pp >= HALO) Acc[ch * ASTR + (pp - HALO)] = h;
    }
  }

  // ---------------- Phase 2: 8 dilated residual blocks ----------------------
  const int start_chunk = (tile == 0) ? (HALO / 16) : 0;  // 16 or 0
  const int nrounds     = ((P_ / 16) - start_chunk) / NW; // 2 or 4

  for (int blk = 0; blk < NBLK; ++blk) {
    const int d = 1 << blk;
    __syncthreads();  // prior Hbuf writes done; W1s free to overwrite

    // stage weights via async global->LDS DMA:
    // W1 rearranged to [tap][o][c] so the A-operand is one b64 LDS load
    const float* gw1 = blk_w1 + blk * (TWO_C * C_ * 2);
    for (int i = tid; i < 16384; i += NTHREADS) {
      int k = i >> 13, rem = i & 8191;      // rem = o*64 + c
      async_g2l_b32(W1s + i, (unsigned)((rem * 2 + k) * 4), gw1);
    }
    const float* gw2 = blk_w2 + blk * (C_ * C_);
    for (int i = tid; i < 4096; i += NTHREADS)
      async_g2l_b32(W2s + i, (unsigned)(i * 4), gw2);
    for (int i = tid; i < 128;  i += NTHREADS) Misc[MB1 + i] = blk_b1[blk * 128 + i];
    for (int i = tid; i < 64;   i += NTHREADS) Misc[MB2 + i] = blk_b2[blk * 64 + i];
    async_wait0();

    v8f stA[4], stB[4];
    int chunkA = -1, chunkB = -1;

    for (int r = 0; r < nrounds; ++r) {
      __syncthreads();                 // all reads of round r-1 complete
      if ((r & 1) == 0) { FLUSH(stA, chunkA); } else { FLUSH(stB, chunkB); }

      int chunk = start_chunk + r * NW + wv;       // wave-uniform
      if (chunk * 16 < P_) {
        const int colp = chunk * 16 + ln;          // B/C/D column -> position
        v8f zacc[4];
        #pragma unroll
        for (int zr = 0; zr < 4; ++zr) zacc[zr] = (v8f){0,0,0,0,0,0,0,0};

        for (int p4 = 0; p4 < 4; ++p4) {           // 16-channel groups of gate
          v8f alo, ahi;
          #pragma unroll
          for (int rr = 0; rr < 8; ++rr) {
            alo[rr] = Misc[MB1 + p4 * 16 + rr + 8 * hf];
            ahi[rr] = Misc[MB1 + 64 + p4 * 16 + rr + 8 * hf];
          }
          #pragma unroll
          for (int tap = 0; tap < 2; ++tap) {      // tap0: t-d, tap1: t
            int psrc = colp - (tap == 0 ? d : 0);
            if (psrc < 0) psrc = 0;                // only garbage region reads this
            const float* wb = W1s + tap * 8192;
            for (int kt = 0; kt < 16; ++kt) {
              int c0 = kt * 4 + 2 * hf;
              v2f bv; bv.x = Hbuf[c0 * HSTR + psrc];
                      bv.y = Hbuf[(c0 + 1) * HSTR + psrc];
              v2f awl = *(const v2f*)(wb + (p4 * 16 + ln) * 64 + c0);
              v2f awh = *(const v2f*)(wb + (64 + p4 * 16 + ln) * 64 + c0);
              alo = wmma4(awl, bv, alo);
              ahi = wmma4(awh, bv, ahi);
            }
          }
          // gate: tanh(a_lo) * sigmoid(a_hi)   (V_TANH_F32 + V_RCP_F32)
          v8f g;
          #pragma unroll
          for (int rr = 0; rr < 8; ++rr)
            g[rr] = HTANH(alo[rr]) * fsigmoid(ahi[rr]);

          // 1x1 conv partial: convert g (C/D layout) -> B layout via half-wave swap
          #pragma unroll
          for (int kb = 0; kb < 4; ++kb) {
            int base = (kb & 1) * 4;
            float pl0 = g[base], pl1 = g[base + 1];
            float ph0 = g[base + 2], ph1 = g[base + 3];
            float s0, s1;
            if (kb < 2) { s0 = sx16(ph0); s1 = sx16(ph1); }
            else        { s0 = sx16(pl0); s1 = sx16(pl1); }
            v2f bg;
            if (kb < 2) { bg.x = hf ? s0 : pl0; bg.y = hf ? s1 : pl1; }
            else        { bg.x = hf ? ph0 : s0; bg.y = hf ? ph1 : s1; }
            int ci = p4 * 16 + kb * 4 + 2 * hf;
            #pragma unroll
            for (int zr = 0; zr < 4; ++zr) {
              v2f aw = *(const v2f*)(W2s + (zr * 16 + ln) * 64 + ci);
              zacc[zr] = wmma4(aw, bg, zacc[zr]);
            }
          }
        }
        // residual + bias (read of OLD h: still in compute phase)
        #pragma unroll
        for (int zr = 0; zr < 4; ++zr) {
          #pragma unroll
          for (int rr = 0; rr < 8; ++rr) {
            int ch = zr * 16 + rr + 8 * hf;
            zacc[zr][rr] += Hbuf[ch * HSTR + colp] + Misc[MB2 + ch];
          }
        }
        // stash: writes lag reads by 2 rounds (256 positions >= 2*d_max)
        if ((r & 1) == 0) {
          #pragma unroll
          for (int zr = 0; zr < 4; ++zr) stA[zr] = zacc[zr];
          chunkA = chunk;
        } else {
          #pragma unroll
          for (int zr = 0; zr < 4; ++zr) stB[zr] = zacc[zr];
          chunkB = chunk;
        }
      }
    }
    __syncthreads();       // all reads of the last rounds complete
    FLUSH(stA, chunkA);
    FLUSH(stB, chunkB);
  }

  // ---------------- Phase 3: mid 1x1 (64 -> 128) via WMMA -------------------
  __syncthreads();         // final flush (Acc) visible; W1s free
  for (int i = tid; i < 8192; i += NTHREADS)
    async_g2l_b32(W1s + i, (unsigned)(i * 4), w_mid);   // [o=128][c=64]
  async_wait0();
  __syncthreads();

  float* Z = Hbuf;         // reuse: [128][ASTR]
  {
    const float inv9 = 1.0f / 9.0f;
    for (int r = 0; r < 2; ++r) {
      int chunk = r * NW + wv;          // 16 output chunks
      int colp  = chunk * 16 + ln;      // 0..255 (output-local position)
      #pragma unroll
      for (int zr = 0; zr < 8; ++zr) {
        v8f zt;
        #pragma unroll
        for (int rr = 0; rr < 8; ++rr) zt[rr] = Misc[MBMID + zr * 16 + rr + 8 * hf];
        for (int kt = 0; kt < 16; ++kt) {
          int c0 = kt * 4 + 2 * hf;
          float a0 = Acc[c0 * ASTR + colp];
          float a1 = Acc[(c0 + 1) * ASTR + colp];
          v2f bv; bv.x = (a0 > 0.0f ? a0 : 0.0f) * inv9;   // relu(acc/9)
                  bv.y = (a1 > 0.0f ? a1 : 0.0f) * inv9;
          v2f aw = *(const v2f*)(W1s + (zr * 16 + ln) * 64 + c0);
          zt = wmma4(aw, bv, zt);
        }
        #pragma unroll
        for (int rr = 0; rr < 8; ++rr)
          Z[(zr * 16 + rr + 8 * hf) * ASTR + colp] = zt[rr];
      }
    }
  }
  __syncthreads();

  // ---------------- Phase 4: heads (pi/mu/sigma), one position per thread ---
  {
    int pos = tid;                       // 0..255
    int t   = t0 + pos;
    float api[4], amu[3], asg[3];
    #pragma unroll
    for (int j = 0; j < 4; ++j) api[j] = Misc[MBPI + j];
    #pragma unroll
    for (int j = 0; j < 3; ++j) { amu[j] = Misc[MBMU + j]; asg[j] = Misc[MBSG + j]; }
    for (int c = 0; c < 128; ++c) {
      float zv = Z[c * ASTR + pos];
      #pragma unroll
      for (int j = 0; j < 4; ++j) api[j] += Misc[MWPI + j * 128 + c] * zv;
      #pragma unroll
      for (int j = 0; j < 3; ++j) {
        amu[j] += Misc[MWMU + j * 128 + c] * zv;
        asg[j] += Misc[MWSG + j * 128 + c] * zv;
      }
    }
    // softmax over 4
    float m = api[0];
    #pragma unroll
    for (int j = 1; j < 4; ++j) m = fmaxf(m, api[j]);
    float e[4], s = 0.0f;
    #pragma unroll
    for (int j = 0; j < 4; ++j) { e[j] = __expf(api[j] - m); s += e[j]; }
    float rs = __builtin_amdgcn_rcpf(s);
    #pragma unroll
    for (int j = 0; j < 4; ++j)
      out[(size_t)(b * 4 + j) * T_FULL + t] = e[j] * rs;
    #pragma unroll
    for (int j = 0; j < 3; ++j) {
      out[OFF_MU + (size_t)(b * 3 + j) * T_FULL + t] = 2.0f * HTANH(amu[j] * 0.5f);
      float v = asg[j];
      out[OFF_SG + (size_t)(b * 3 + j) * T_FULL + t] =
          (v > 20.0f) ? v : log1pf(__expf(v));
    }
  }
}

extern "C" void kernel_launch(void* const* d_in, const int* in_sizes, int n_in,
                              void* d_out, int out_size, void* d_ws, size_t ws_size,
                              hipStream_t stream) {
  (void)in_sizes; (void)n_in; (void)d_ws; (void)ws_size; (void)out_size;
  const float* x      = (const float*)d_in[0];
  const float* w_in   = (const float*)d_in[1];
  const float* b_in   = (const float*)d_in[2];
  const float* blk_w1 = (const float*)d_in[3];
  const float* blk_b1 = (const float*)d_in[4];
  const float* blk_w2 = (const float*)d_in[5];
  const float* blk_b2 = (const float*)d_in[6];
  const float* w_mid  = (const float*)d_in[7];
  const float* b_mid  = (const float*)d_in[8];
  const float* w_pi   = (const float*)d_in[9];
  const float* b_pi   = (const float*)d_in[10];
  const float* w_mu   = (const float*)d_in[11];
  const float* b_mu   = (const float*)d_in[12];
  const float* w_sg   = (const float*)d_in[13];
  const float* b_sg   = (const float*)d_in[14];
  float* out = (float*)d_out;

  dim3 grid(BATCH * NTILES);                 // 1024 workgroups: (batch, time tile)
  size_t shmem = (size_t)SMEM_FLOATS * sizeof(float);  // 292,864 B of LDS
  Mooncake_wavenet_kernel<<<grid, NTHREADS, shmem, stream>>>(
      x, w_in, b_in, blk_w1, blk_b1, blk_w2, blk_b2,
      w_mid, b_mid, w_pi, b_pi, w_mu, b_mu, w_sg, b_sg, out);
}